// SelfGeneratingRecognitionTensor_1142461301146
// MI455X (gfx1250) — compile-verified
//
#include <hip/hip_runtime.h>
#include <hip/hip_bf16.h>
#include <stdint.h>

#define NN 2048
#define NNLL ((long long)NN * (long long)NN)
#define TAYLOR_K 8
#define KC 8           // K-chunk staged in LDS
#define LSTR 12        // padded LDS stride (bank-conflict-free, 16B-align OK)

typedef float v2f __attribute__((ext_vector_type(2)));
typedef float v4f __attribute__((ext_vector_type(4)));
typedef float v8f __attribute__((ext_vector_type(8)));

// ---------------------------------------------------------------------------
// Scalar helpers
// ---------------------------------------------------------------------------
__global__ void zero_scalars_kernel(double* s) {
    if (threadIdx.x < 2) s[threadIdx.x] = 0.0;
}

// Frobenius-norm^2 accumulation in fp64 (matches fp64 reference for scalars).
__global__ __launch_bounds__(256) void reduce_sumsq_kernel(
    const float* __restrict__ x, long long n, double* __restrict__ out) {
    __shared__ double sm[256];
    double acc = 0.0;
    for (long long i = (long long)blockIdx.x * blockDim.x + threadIdx.x; i < n;
         i += (long long)gridDim.x * blockDim.x) {
        double v = (double)x[i];
        acc += v * v;
    }
    sm[threadIdx.x] = acc;
    __syncthreads();
    for (int s = 128; s > 0; s >>= 1) {
        if ((int)threadIdx.x < s) sm[threadIdx.x] += sm[threadIdx.x + s];
        __syncthreads();
    }
    if (threadIdx.x == 0) atomicAdd(out, sm[0]);
}

// ---------------------------------------------------------------------------
// A = -i*dt*H + dt*rate*D ,  H = 0.5(R + R^H), D = 0.5(R - R^H),  R = X + iY:
//   A_real = dt*0.5*( (Y - Y^T) + rate*(X - X^T) )
//   A_imag = dt*0.5*( -(X + X^T) + rate*(Y + Y^T) )
// ---------------------------------------------------------------------------
__global__ __launch_bounds__(256) void build_A_kernel(
    const float* __restrict__ Rr, const float* __restrict__ Ri,
    const float* __restrict__ rate_p, float* __restrict__ Ar,
    float* __restrict__ Ai) {
    long long idx = (long long)blockIdx.x * 256 + threadIdx.x;
    if (idx >= NNLL) return;
    int i = (int)(idx >> 11);
    int j = (int)(idx & (NN - 1));
    long long tidx = (long long)j * NN + i;
    const float dt = 0.01f;
    float rate = rate_p[0];
    float x = Rr[idx], xt = Rr[tidx];
    float y = Ri[idx], yt = Ri[tidx];
    Ar[idx] = dt * 0.5f * ((y - yt) + rate * (x - xt));
    Ai[idx] = dt * 0.5f * (-(x + xt) + rate * (y + yt));
}

// P = I + A * invK   (innermost Horner term)
__global__ __launch_bounds__(256) void init_P_kernel(
    const float* __restrict__ Ar, const float* __restrict__ Ai,
    float* __restrict__ Pr, float* __restrict__ Pi, float invK) {
    long long idx = (long long)blockIdx.x * 256 + threadIdx.x;
    if (idx >= NNLL) return;
    int i = (int)(idx >> 11);
    int j = (int)(idx & (NN - 1));
    Pr[idx] = Ar[idx] * invK + ((i == j) ? 1.0f : 0.0f);
    Pi[idx] = Ai[idx] * invK;
}

// 32-bit DS offset of a __shared__ element (low 32 bits of the address).
__device__ __forceinline__ unsigned lds_off(const void* p) {
    return (unsigned)(uintptr_t)p;
}

// ---------------------------------------------------------------------------
// Complex GEMM via V_WMMA_F32_16X16X4_F32 with ASYNC global->LDS staging
// (GLOBAL_LOAD_ASYNC_TO_LDS_*, ASYNCcnt) and double buffering.
//   C = scale * (A @ B)    (CONJB=false)   or   C = scale * (A @ B^H) (CONJB=true)
// optionally += I.
// Block tile 128x128 (8 waves, 4 row x 2 col), wave tile 32x64 complex
// (2x4 subtiles of 16x16; 4 WMMAs per subtile per k-step).
// LDS: A tile 128xKC row-major (stride LSTR); B tile stored N-major
// (col*LSTR + k) so fragment reads are contiguous ds_load_b64. For the
// normal-B operand the k-major -> N-major transpose happens *inside* the
// async copy (per-lane global and LDS addresses differ). Conjugation for
// CONJB is folded into the WMMA wiring (one negated A fragment either way):
//   normal: Cr += Ar*Br + (-Ai)*Bi ; Ci += Ar*Bi + Ai*Br
//   conjB : Cr += Ar*Br +  Ai *Bi ; Ci += (-Ar)*Bi + Ai*Br   (raw Bi in LDS)
// Stride 12 floats => 16 half-wave lanes on 16 distinct banks, 8B/16B aligned.
// ---------------------------------------------------------------------------
template <bool CONJB>
__global__ __launch_bounds__(256) void cgemm16_kernel(
    const float* __restrict__ Ar, const float* __restrict__ Ai,
    const float* __restrict__ Br, const float* __restrict__ Bi,
    float* __restrict__ Cr, float* __restrict__ Ci, float scale, int addIdent) {
    __shared__ float sA[2][2][128 * LSTR];  // [buf][re/im][row*LSTR + k]
    __shared__ float sB[2][2][128 * LSTR];  // [buf][re/im][col*LSTR + k]

    const int t    = threadIdx.x;
    const int lane = t & 31;
    const int wid  = t >> 5;
    const int wm   = wid & 3;   // 0..3 -> 32-row band
    const int wn   = wid >> 2;  // 0..1 -> 64-col band
    const int rowBase = blockIdx.y * 128;
    const int colBase = blockIdx.x * 128;
    const int lr = lane & 15;
    const int kk = (lane >> 4) << 1;  // 0 or 2

    // staging thread mapping
    const int rowA = t >> 1;        // 0..127
    const int qA   = (t & 1) << 2;  // 0 or 4
    const int cB   = t & 127;       // normal-B: column
    const int kB   = (t >> 7) << 2; // normal-B: k group (0 or 4)

    v8f accr[2][4] = {};
    v8f acci[2][4] = {};

    // Issue async L2->LDS copies for chunk starting at k=kc into buffer `buf`.
    auto stage_issue = [&](int kc, int buf) {
        const float* gar = Ar + (long long)(rowBase + rowA) * NN + kc + qA;
        const float* gai = Ai + (long long)(rowBase + rowA) * NN + kc + qA;
        unsigned la0 = lds_off(&sA[buf][0][rowA * LSTR + qA]);
        unsigned la1 = lds_off(&sA[buf][1][rowA * LSTR + qA]);
        asm volatile("global_load_async_to_lds_b128 %0, %1, off"
                     :: "v"(la0), "v"(gar) : "memory");
        asm volatile("global_load_async_to_lds_b128 %0, %1, off"
                     :: "v"(la1), "v"(gai) : "memory");
        if (CONJB) {
            // B^H(k,c) from row c of B: contiguous 16B both sides
            const float* gbr = Br + (long long)(colBase + rowA) * NN + kc + qA;
            const float* gbi = Bi + (long long)(colBase + rowA) * NN + kc + qA;
            unsigned lb0 = lds_off(&sB[buf][0][rowA * LSTR + qA]);
            unsigned lb1 = lds_off(&sB[buf][1][rowA * LSTR + qA]);
            asm volatile("global_load_async_to_lds_b128 %0, %1, off"
                         :: "v"(lb0), "v"(gbr) : "memory");
            asm volatile("global_load_async_to_lds_b128 %0, %1, off"
                         :: "v"(lb1), "v"(gbi) : "memory");
        } else {
            // k-major global -> N-major LDS: transpose inside the async copy
#pragma unroll
            for (int i = 0; i < 4; ++i) {
                const float* gbr = Br + (long long)(kc + kB + i) * NN + colBase + cB;
                const float* gbi = Bi + (long long)(kc + kB + i) * NN + colBase + cB;
                unsigned lb0 = lds_off(&sB[buf][0][cB * LSTR + kB + i]);
                unsigned lb1 = lds_off(&sB[buf][1][cB * LSTR + kB + i]);
                asm volatile("global_load_async_to_lds_b32 %0, %1, off"
                             :: "v"(lb0), "v"(gbr) : "memory");
                asm volatile("global_load_async_to_lds_b32 %0, %1, off"
                             :: "v"(lb1), "v"(gbi) : "memory");
            }
        }
    };

    stage_issue(0, 0);
    asm volatile("s_wait_asynccnt 0x0" ::: "memory");
    __syncthreads();

    const int nchunks = NN / KC;
    for (int ch = 0; ch < nchunks; ++ch) {
        const int buf = ch & 1;
        if (ch + 1 < nchunks) stage_issue((ch + 1) * KC, buf ^ 1);  // overlap

#pragma unroll
        for (int k0 = 0; k0 < KC; k0 += 4) {
            v2f afr[2], afi[2], nA[2];
#pragma unroll
            for (int tr = 0; tr < 2; ++tr) {
                const int off = (wm * 32 + tr * 16 + lr) * LSTR + k0 + kk;
                afr[tr] = *(const v2f*)&sA[buf][0][off];
                afi[tr] = *(const v2f*)&sA[buf][1][off];
                nA[tr]  = CONJB ? -afr[tr] : -afi[tr];  // one negation either way
            }
            v2f bfr[4], bfi[4];
#pragma unroll
            for (int tc = 0; tc < 4; ++tc) {
                const int off = (wn * 64 + tc * 16 + lr) * LSTR + k0 + kk;
                bfr[tc] = *(const v2f*)&sB[buf][0][off];
                bfi[tc] = *(const v2f*)&sB[buf][1][off];
            }
#pragma unroll
            for (int tr = 0; tr < 2; ++tr) {
#pragma unroll
                for (int tc = 0; tc < 4; ++tc) {
                    accr[tr][tc] = __builtin_amdgcn_wmma_f32_16x16x4_f32(
                        false, afr[tr], false, bfr[tc], (short)0, accr[tr][tc], false, false);
                    accr[tr][tc] = __builtin_amdgcn_wmma_f32_16x16x4_f32(
                        false, CONJB ? afi[tr] : nA[tr], false, bfi[tc], (short)0,
                        accr[tr][tc], false, false);
                    acci[tr][tc] = __builtin_amdgcn_wmma_f32_16x16x4_f32(
                        false, CONJB ? nA[tr] : afr[tr], false, bfi[tc], (short)0,
                        acci[tr][tc], false, false);
                    acci[tr][tc] = __builtin_amdgcn_wmma_f32_16x16x4_f32(
                        false, afi[tr], false, bfr[tc], (short)0, acci[tr][tc], false, false);
                }
            }
        }

        // all async copies (next chunk) must land before anyone reads them
        asm volatile("s_wait_asynccnt 0x0" ::: "memory");
        __syncthreads();
    }

    // ---- store (coalesced along N across lanes) ----
    const int mo = (lane >> 4) << 3;  // 0 or 8
#pragma unroll
    for (int tr = 0; tr < 2; ++tr) {
#pragma unroll
        for (int tc = 0; tc < 4; ++tc) {
            const int c = colBase + wn * 64 + tc * 16 + lr;
#pragma unroll
            for (int v = 0; v < 8; ++v) {
                const int r = rowBase + wm * 32 + tr * 16 + mo + v;
                float vr = scale * accr[tr][tc][v];
                float vi = scale * acci[tr][tc][v];
                if (addIdent && r == c) vr += 1.0f;
                Cr[(long long)r * NN + c] = vr;
                Ci[(long long)r * NN + c] = vi;
            }
        }
    }
}

// ---------------------------------------------------------------------------
// Trace-normalize new_R in d_out and emit the awareness scalar.
// scal[0] = sum|R|^2 (original), scal[1] = sum|new_R|^2 (unscaled)
// ---------------------------------------------------------------------------
__global__ __launch_bounds__(256) void finalize_kernel(
    float* __restrict__ out, long long m, const double* __restrict__ scal) {
    long long idx = (long long)blockIdx.x * 256 + threadIdx.x;
    double nrm = sqrt(scal[1]);
    float scale = (nrm > 1e-10) ? (float)(sqrt((double)NN) / nrm) : 1.0f;
    if (idx < 2 * m) out[idx] *= scale;
    if (idx == 0) {
        // t=0: temporal_modulation = cos(0) = 1 -> factor 1.1
        out[2 * m] = (float)(1.1 * scal[0] / (double)NN);
    }
}

// ---------------------------------------------------------------------------
extern "C" void kernel_launch(void* const* d_in, const int* in_sizes, int n_in,
                              void* d_out, int out_size, void* d_ws, size_t ws_size,
                              hipStream_t stream) {
    (void)in_sizes; (void)n_in; (void)out_size; (void)ws_size;
    const float* Rr   = (const float*)d_in[0];
    const float* Ri   = (const float*)d_in[1];
    const float* rate = (const float*)d_in[2];

    const size_t m = (size_t)NN * NN;
    float* Ar = (float*)d_ws;
    float* Ai = Ar + m;
    float* Pr = Ai + m;
    float* Pi = Pr + m;
    float* Qr = Pi + m;
    float* Qi = Qr + m;
    double* scal = (double*)(Qi + m);  // 96MB offset -> 8B aligned

    float* outR = (float*)d_out;
    float* outI = outR + m;

    const int ew = (int)((m + 255) / 256);
    dim3 g(NN / 128, NN / 128), b(256);

    // scalars must be re-zeroed every call (graph replays do not re-poison)
    zero_scalars_kernel<<<1, 32, 0, stream>>>(scal);

    // awareness: sum |R|^2
    reduce_sumsq_kernel<<<512, 256, 0, stream>>>(Rr, (long long)m, scal);
    reduce_sumsq_kernel<<<512, 256, 0, stream>>>(Ri, (long long)m, scal);

    // generator A and innermost Horner term
    build_A_kernel<<<ew, 256, 0, stream>>>(Rr, Ri, rate, Ar, Ai);
    init_P_kernel<<<ew, 256, 0, stream>>>(Ar, Ai, Pr, Pi, 1.0f / TAYLOR_K);

    // Horner: P <- I + (A/k) * P ,  k = K-1 .. 1  => P = expm(A) to O(A^9)
    for (int k = TAYLOR_K - 1; k >= 1; --k) {
        cgemm16_kernel<false><<<g, b, 0, stream>>>(Ar, Ai, Pr, Pi, Qr, Qi,
                                                   1.0f / (float)k, 1);
        float* t;
        t = Pr; Pr = Qr; Qr = t;
        t = Pi; Pi = Qi; Qi = t;
    }

    // T = R @ U^H  (A buffers are free now; conj folded into WMMA wiring)
    cgemm16_kernel<true><<<g, b, 0, stream>>>(Rr, Ri, Pr, Pi, Ar, Ai, 1.0f, 0);
    // new_R = U @ T  -> d_out (unscaled): [0,m) real, [m,2m) imag
    cgemm16_kernel<false><<<g, b, 0, stream>>>(Pr, Pi, Ar, Ai, outR, outI, 1.0f, 0);

    // trace normalization + awareness scalar
    reduce_sumsq_kernel<<<512, 256, 0, stream>>>(outR, (long long)(2 * m), scal + 1);
    finalize_kernel<<<(int)((2 * m + 255) / 256), 256, 0, stream>>>(
        outR, (long long)m, scal);
}